// PGAHead_33200097198779
// MI455X (gfx1250) — compile-verified
//
#include <hip/hip_runtime.h>
#include <hip/hip_bf16.h>

typedef float v2f __attribute__((ext_vector_type(2)));
typedef float v8f __attribute__((ext_vector_type(8)));

#define Bsz 2048
#define Dsz 512
#define Lsz 3
#define Psz 768
#define TOPKK 64

#if defined(__gfx1250__) && __has_builtin(__builtin_amdgcn_global_load_async_to_lds_b128)
#define HAVE_ASYNC_LDS 1
#else
#define HAVE_ASYNC_LDS 0
#endif

// ---------------------------------------------------------------------------
// WMMA f32 GEMM: C[M,N] = A[M,K] * op(B); op(B)=B (NN, B is K x N) or B^T (NT,
// B stored N x K row-major). M,N multiples of 128; K multiple of 16.
// Block = 256 threads = 8 waves. Block tile 128x128. Wave (wr,wc) in 4x2 grid
// owns a 32x64 region = 2x4 WMMA 16x16 accumulators.
// LDS: A tile staged [128][20] (16B-aligned rows, conflict-free col access);
// B tile staged [16][128] (NN) or [128][20] (NT). Tiles staged with
// GLOBAL_LOAD_ASYNC_TO_LDS_B128 when available.
// ---------------------------------------------------------------------------
__global__ __launch_bounds__(256) void wmma_gemm_f32(
    const float* __restrict__ A, const float* __restrict__ B,
    float* __restrict__ C, int M, int N, int K,
    int lda, int ldb, int ldc, int transB)
{
    __shared__ float sA[128 * 20];
    __shared__ float sB[128 * 20];

    const int t    = threadIdx.x;
    const int lane = t & 31;
    const int wave = t >> 5;
    const int wr   = wave >> 1;     // wave row (0..3) -> 32 rows each
    const int wc   = wave & 1;      // wave col (0..1) -> 64 cols each
    const int m0   = blockIdx.y * 128;
    const int n0   = blockIdx.x * 128;
    const int half = lane >> 4;     // 0: lanes 0-15, 1: lanes 16-31
    const int l    = lane & 15;

    v8f acc[2][4];
#pragma unroll
    for (int i = 0; i < 2; ++i)
#pragma unroll
        for (int j = 0; j < 4; ++j)
            acc[i][j] = (v8f){};

    for (int k0 = 0; k0 < K; k0 += 16) {
#if HAVE_ASYNC_LDS
        {
            typedef int v4i __attribute__((vector_size(16)));
            typedef __attribute__((address_space(1))) v4i g_v4i;
            typedef __attribute__((address_space(3))) v4i l_v4i;
            // A tile: 128 rows x 16 cols, 16B chunks; 2 issues of 256 lanes.
#pragma unroll
            for (int issue = 0; issue < 2; ++issue) {
                int c = t + issue * 256;
                int row = c >> 2, col4 = c & 3;
                __builtin_amdgcn_global_load_async_to_lds_b128(
                    (g_v4i*)(A + (size_t)(m0 + row) * lda + k0 + col4 * 4),
                    (l_v4i*)(sA + row * 20 + col4 * 4), 0, 0);
            }
            if (transB) {
                // B tile as [n][k]: 128 rows x 16 cols
#pragma unroll
                for (int issue = 0; issue < 2; ++issue) {
                    int c = t + issue * 256;
                    int row = c >> 2, col4 = c & 3;
                    __builtin_amdgcn_global_load_async_to_lds_b128(
                        (g_v4i*)(B + (size_t)(n0 + row) * ldb + k0 + col4 * 4),
                        (l_v4i*)(sB + row * 20 + col4 * 4), 0, 0);
                }
            } else {
                // B tile as [k][n]: 16 rows x 128 cols
#pragma unroll
                for (int issue = 0; issue < 2; ++issue) {
                    int c = t + issue * 256;
                    int row = c >> 5, col32 = c & 31;
                    __builtin_amdgcn_global_load_async_to_lds_b128(
                        (g_v4i*)(B + (size_t)(k0 + row) * ldb + n0 + col32 * 4),
                        (l_v4i*)(sB + row * 128 + col32 * 4), 0, 0);
                }
            }
#if __has_builtin(__builtin_amdgcn_s_wait_asynccnt)
            __builtin_amdgcn_s_wait_asynccnt(0);
#else
            asm volatile("s_wait_asynccnt 0x0" ::: "memory");
#endif
        }
#else
        // Fallback staging through VGPRs
#pragma unroll
        for (int r = 0; r < 8; ++r) {
            int idx = t + r * 256;
            int am = idx >> 4, ak = idx & 15;
            sA[am * 20 + ak] = A[(size_t)(m0 + am) * lda + (k0 + ak)];
        }
        if (transB) {
#pragma unroll
            for (int r = 0; r < 8; ++r) {
                int idx = t + r * 256;
                int bn = idx >> 4, bk = idx & 15;
                sB[bn * 20 + bk] = B[(size_t)(n0 + bn) * ldb + (k0 + bk)];
            }
        } else {
#pragma unroll
            for (int r = 0; r < 8; ++r) {
                int idx = t + r * 256;
                int bk = idx >> 7, bn = idx & 127;
                sB[bk * 128 + bn] = B[(size_t)(k0 + bk) * ldb + (n0 + bn)];
            }
        }
#endif
        __syncthreads();

#pragma unroll
        for (int kk = 0; kk < 4; ++kk) {
            const int kb = kk * 4 + half * 2;   // lanes 0-15: K=0,1; 16-31: K=2,3
            v2f afrag[2], bfrag[4];
#pragma unroll
            for (int i = 0; i < 2; ++i) {
                int m = wr * 32 + i * 16 + l;
                afrag[i][0] = sA[m * 20 + kb];
                afrag[i][1] = sA[m * 20 + kb + 1];
            }
            if (transB) {
#pragma unroll
                for (int j = 0; j < 4; ++j) {
                    int n = wc * 64 + j * 16 + l;
                    bfrag[j][0] = sB[n * 20 + kb];
                    bfrag[j][1] = sB[n * 20 + kb + 1];
                }
            } else {
#pragma unroll
                for (int j = 0; j < 4; ++j) {
                    int n = wc * 64 + j * 16 + l;
                    bfrag[j][0] = sB[kb * 128 + n];
                    bfrag[j][1] = sB[(kb + 1) * 128 + n];
                }
            }
#pragma unroll
            for (int i = 0; i < 2; ++i)
#pragma unroll
                for (int j = 0; j < 4; ++j)
                    acc[i][j] = __builtin_amdgcn_wmma_f32_16x16x4_f32(
                        false, afrag[i], false, bfrag[j], (short)0, acc[i][j], false, false);
        }
        __syncthreads();
    }

    // C/D layout: VGPR r -> (M = r + 8*half, N = l)
#pragma unroll
    for (int i = 0; i < 2; ++i) {
        const int cm0 = m0 + wr * 32 + i * 16 + half * 8;
#pragma unroll
        for (int j = 0; j < 4; ++j) {
            const int cn = n0 + wc * 64 + j * 16 + l;
#pragma unroll
            for (int r = 0; r < 8; ++r)
                C[(size_t)(cm0 + r) * ldc + cn] = acc[i][j][r];
        }
    }
}

// ---------------------------------------------------------------------------
// Row L2 normalize: out[r,:] = in[r,:] / max(||in[r,:]||, 1e-12)
// ---------------------------------------------------------------------------
__global__ void l2norm_rows_kernel(const float* __restrict__ in,
                                   float* __restrict__ out, int C)
{
    const int row = blockIdx.x;
    const int t = threadIdx.x;
    __shared__ float red[256];
    __shared__ float invn;
    const float* p = in + (size_t)row * C;
    float ss = 0.f;
    for (int c = t; c < C; c += 256) { float v = p[c]; ss += v * v; }
    red[t] = ss; __syncthreads();
    for (int s = 128; s > 0; s >>= 1) { if (t < s) red[t] += red[t + s]; __syncthreads(); }
    if (t == 0) invn = 1.0f / fmaxf(sqrtf(red[0]), 1e-12f);
    __syncthreads();
    float* q = out + (size_t)row * C;
    for (int c = t; c < C; c += 256) q[c] = p[c] * invn;
}

__global__ void clip_kernel(float* __restrict__ S, size_t n)
{
    size_t stride = (size_t)gridDim.x * blockDim.x;
    for (size_t i = (size_t)blockIdx.x * blockDim.x + threadIdx.x; i < n; i += stride)
        S[i] = fminf(fmaxf(S[i], -1.0f + 1e-8f), 1.0f - 1e-8f);
}

// ---------------------------------------------------------------------------
// Top-64 mask per row over masked similarities (matches jax top_k tie-break:
// lowest index first). Output: bitmask, 64 words per row.
// ---------------------------------------------------------------------------
__device__ __forceinline__ unsigned f2ord(float x)
{
    unsigned u = __float_as_uint(x);
    return (u & 0x80000000u) ? ~u : (u | 0x80000000u);
}

__global__ void topk_mask_kernel(const float* __restrict__ S,
                                 const int* __restrict__ labels,
                                 unsigned* __restrict__ maskBits, int wantSame)
{
    const int row = blockIdx.x;
    const int t = threadIdx.x;   // 256
    __shared__ unsigned red[256];
    __shared__ unsigned scanBuf[256];
    __shared__ unsigned words[64];
    const int myLab = labels[row];
    const float* Srow = S + (size_t)row * Bsz;

    unsigned key[8];
#pragma unroll
    for (int r = 0; r < 8; ++r) {
        int j = t * 8 + r;
        int same = (labels[j] == myLab);
        int bm = wantSame ? same : !same;
        float v = bm ? (Srow[j] - ((j == row) ? 1e9f : 0.0f)) : -1e9f;
        key[r] = f2ord(v);
    }

    // binary search for 64-th largest key: max T with count(key >= T) >= 64
    unsigned lo = 0u, hi = 0xFFFFFFFFu;
    for (int it = 0; it < 32; ++it) {
        unsigned mid = (unsigned)(((unsigned long long)lo + (unsigned long long)hi + 1ull) >> 1);
        unsigned c = 0;
#pragma unroll
        for (int r = 0; r < 8; ++r) c += (key[r] >= mid);
        red[t] = c; __syncthreads();
        for (int s = 128; s > 0; s >>= 1) { if (t < s) red[t] += red[t + s]; __syncthreads(); }
        unsigned total = red[0]; __syncthreads();
        if (total >= (unsigned)TOPKK) lo = mid; else hi = mid - 1u;
    }
    const unsigned thr = lo;

    // count strictly-greater
    unsigned cg = 0;
#pragma unroll
    for (int r = 0; r < 8; ++r) cg += (key[r] > thr);
    red[t] = cg; __syncthreads();
    for (int s = 128; s > 0; s >>= 1) { if (t < s) red[t] += red[t + s]; __syncthreads(); }
    const unsigned n_gt = red[0]; __syncthreads();
    const unsigned tieNeed = (unsigned)TOPKK - n_gt;

    // exclusive prefix of tie counts in index order
    unsigned myTies = 0;
#pragma unroll
    for (int r = 0; r < 8; ++r) myTies += (key[r] == thr);
    scanBuf[t] = myTies; __syncthreads();
    for (int s = 1; s < 256; s <<= 1) {
        unsigned v = (t >= s) ? scanBuf[t - s] : 0u;
        __syncthreads();
        scanBuf[t] += v;
        __syncthreads();
    }
    unsigned tieRank = scanBuf[t] - myTies;

    unsigned byte = 0;
#pragma unroll
    for (int r = 0; r < 8; ++r) {
        bool set = false;
        if (key[r] > thr) set = true;
        else if (key[r] == thr) { set = (tieRank < tieNeed); tieRank++; }
        if (set) byte |= (1u << r);
    }
    if (t < 64) words[t] = 0u;
    __syncthreads();
    atomicOr(&words[t >> 2], byte << ((t & 3) * 8));
    __syncthreads();
    if (t < 64) maskBits[(size_t)row * 64 + t] = words[t];
}

// ---------------------------------------------------------------------------
// A = alpha*relu(S)*Msym_intra + beta*relu(S)*Msym_inter + 1e-6*I (in place),
// plus row sums into deg.
// ---------------------------------------------------------------------------
__global__ void build_A_kernel(float* __restrict__ S,
                               const unsigned* __restrict__ mIntra,
                               const unsigned* __restrict__ mInter,
                               float* __restrict__ deg, float alpha, float beta)
{
    const int i = blockIdx.x;
    const int t = threadIdx.x;
    __shared__ float red[256];
    float sum = 0.f;
#pragma unroll
    for (int r = 0; r < 8; ++r) {
        int j = t + r * 256;
        unsigned wi = mIntra[(size_t)i * 64 + (j >> 5)];
        unsigned wj = mIntra[(size_t)j * 64 + (i >> 5)];
        int Mi = (int)(((wi >> (j & 31)) | (wj >> (i & 31))) & 1u);
        unsigned vi = mInter[(size_t)i * 64 + (j >> 5)];
        unsigned vj = mInter[(size_t)j * 64 + (i >> 5)];
        int Me = (int)(((vi >> (j & 31)) | (vj >> (i & 31))) & 1u);
        float s = S[(size_t)i * Bsz + j];
        float sp = fmaxf(s, 0.0f);
        float a = alpha * (Mi ? sp : 0.0f) + beta * (Me ? sp : 0.0f) + ((i == j) ? 1e-6f : 0.0f);
        S[(size_t)i * Bsz + j] = a;
        sum += a;
    }
    red[t] = sum; __syncthreads();
    for (int s = 128; s > 0; s >>= 1) { if (t < s) red[t] += red[t + s]; __syncthreads(); }
    if (t == 0) deg[i] = red[0];
}

__global__ void dinv_kernel(const float* __restrict__ deg, float* __restrict__ dinv)
{
    int i = blockIdx.x * blockDim.x + threadIdx.x;
    if (i < Bsz) dinv[i] = 1.0f / sqrtf(fmaxf(deg[i], 1e-8f));
}

__global__ void normsym_kernel(float* __restrict__ A, const float* __restrict__ dinv)
{
    const size_t n = (size_t)Bsz * Bsz;
    size_t stride = (size_t)gridDim.x * blockDim.x;
    for (size_t idx = (size_t)blockIdx.x * blockDim.x + threadIdx.x; idx < n; idx += stride) {
        int i = (int)(idx >> 11);
        int j = (int)(idx & 2047);
        A[idx] *= dinv[i] * dinv[j];
    }
}

// ---------------------------------------------------------------------------
// BatchNorm over axis 0 (batch stats, biased var) + ReLU. One block per column.
// ---------------------------------------------------------------------------
__global__ void bn_relu_kernel(const float* __restrict__ Zin, float* __restrict__ Zout,
                               const float* __restrict__ gamma, const float* __restrict__ bbeta)
{
    const int c = blockIdx.x;
    const int t = threadIdx.x;
    __shared__ float red[256];
    __shared__ float mu_s, sc_s;
    float s = 0.f;
    for (int r = t; r < Bsz; r += 256) s += Zin[(size_t)r * Dsz + c];
    red[t] = s; __syncthreads();
    for (int q = 128; q > 0; q >>= 1) { if (t < q) red[t] += red[t + q]; __syncthreads(); }
    if (t == 0) mu_s = red[0] * (1.0f / (float)Bsz);
    __syncthreads();
    float v = 0.f;
    for (int r = t; r < Bsz; r += 256) { float d = Zin[(size_t)r * Dsz + c] - mu_s; v += d * d; }
    red[t] = v; __syncthreads();
    for (int q = 128; q > 0; q >>= 1) { if (t < q) red[t] += red[t + q]; __syncthreads(); }
    if (t == 0) sc_s = rsqrtf(red[0] * (1.0f / (float)Bsz) + 1e-5f) * gamma[c];
    __syncthreads();
    const float bb = bbeta[c];
    for (int r = t; r < Bsz; r += 256) {
        float z = (Zin[(size_t)r * Dsz + c] - mu_s) * sc_s + bb;
        Zout[(size_t)r * Dsz + c] = fmaxf(z, 0.0f);
    }
}

__global__ void add_kernel(float* __restrict__ Z, const float* __restrict__ X, size_t n)
{
    size_t stride = (size_t)gridDim.x * blockDim.x;
    for (size_t i = (size_t)blockIdx.x * blockDim.x + threadIdx.x; i < n; i += stride)
        Z[i] += X[i];
}

// ---------------------------------------------------------------------------
// Deterministic two-stage squared-difference reductions.
// ---------------------------------------------------------------------------
__global__ void sqdiff_partial_kernel(const float* __restrict__ a, const float* __restrict__ b,
                                      float* __restrict__ partials, size_t n)
{
    __shared__ float red[256];
    const int t = threadIdx.x;
    size_t stride = (size_t)gridDim.x * blockDim.x;
    float s = 0.f;
    for (size_t i = (size_t)blockIdx.x * blockDim.x + t; i < n; i += stride) {
        float d = a[i] - b[i];
        s += d * d;
    }
    red[t] = s; __syncthreads();
    for (int q = 128; q > 0; q >>= 1) { if (t < q) red[t] += red[t + q]; __syncthreads(); }
    if (t == 0) partials[blockIdx.x] = red[0];
}

__global__ void idea_dinv_kernel(const int* __restrict__ labels, float* __restrict__ dinvI)
{
    int i = blockIdx.x * blockDim.x + threadIdx.x;
    if (i >= Bsz) return;
    int lab = labels[i];
    int c = 0;
    for (int j = 0; j < Bsz; ++j) c += (labels[j] == lab);
    float sum = 1.0f + 0.98f * (float)(c - 1) + 0.03f * (float)(Bsz - c);
    dinvI[i] = 1.0f / sqrtf(fmaxf(sum, 1e-8f));
}

__global__ void idea_partial_kernel(const float* __restrict__ Klast,
                                    const int* __restrict__ labels,
                                    const float* __restrict__ dinvI,
                                    float* __restrict__ partials)
{
    __shared__ float red[256];
    const int t = threadIdx.x;
    const size_t n = (size_t)Bsz * Bsz;
    size_t stride = (size_t)gridDim.x * blockDim.x;
    float s = 0.f;
    for (size_t idx = (size_t)blockIdx.x * blockDim.x + t; idx < n; idx += stride) {
        int i = (int)(idx >> 11);
        int j = (int)(idx & 2047);
        float aij = (labels[i] == labels[j]) ? ((i == j) ? 1.0f : 0.98f) : 0.03f;
        float kid = aij * dinvI[i] * dinvI[j];
        float d = Klast[idx] - kid;
        s += d * d;
    }
    red[t] = s; __syncthreads();
    for (int q = 128; q > 0; q >>= 1) { if (t < q) red[t] += red[t + q]; __syncthreads(); }
    if (t == 0) partials[blockIdx.x] = red[0];
}

__global__ void finalize_add_kernel(const float* __restrict__ partials, int np,
                                    float* __restrict__ scal, float scale)
{
    __shared__ float red[256];
    const int t = threadIdx.x;
    float s = 0.f;
    for (int i = t; i < np; i += 256) s += partials[i];
    red[t] = s; __syncthreads();
    for (int q = 128; q > 0; q >>= 1) { if (t < q) red[t] += red[t + q]; __syncthreads(); }
    if (t == 0) scal[0] += red[0] * scale;
}

__global__ void init_scalars_kernel(float* __restrict__ scal)
{
    if (threadIdx.x < 8) scal[threadIdx.x] = 0.0f;
}

__global__ void write_losses_kernel(const float* __restrict__ scal, float* __restrict__ outTail)
{
    if (threadIdx.x == 0) {
        float lk = scal[0], lz = scal[1], li = scal[2];
        outTail[0] = lk;
        outTail[1] = lz;
        outTail[2] = li;
        outTail[3] = 0.5f * lk + 1.0f * lz + 1.0f * li;
    }
}

// ---------------------------------------------------------------------------

static inline void gemm(const float* A, const float* B, float* C,
                        int M, int N, int K, int lda, int ldb, int ldc,
                        int transB, hipStream_t s)
{
    dim3 g(N / 128, M / 128);
    wmma_gemm_f32<<<g, 256, 0, s>>>(A, B, C, M, N, K, lda, ldb, ldc, transB);
}

extern "C" void kernel_launch(void* const* d_in, const int* in_sizes, int n_in,
                              void* d_out, int out_size, void* d_ws, size_t ws_size,
                              hipStream_t stream)
{
    const float* feats  = (const float*)d_in[0];   // [3,2048,512]
    const int*   labels = (const int*)  d_in[1];   // [2048]
    const float* fc1    = (const float*)d_in[2];   // [3,512,512]
    const float* fc2    = (const float*)d_in[3];   // [3,512,512]
    const float* gamma  = (const float*)d_in[4];   // [3,512]
    const float* bbeta  = (const float*)d_in[5];   // [3,512]
    const float* proj   = (const float*)d_in[6];   // [768,512]
    float* Zout = (float*)d_out;                   // [3,2048,512] + 4 scalars

    float* w = (float*)d_ws;
    size_t o = 0;
    auto alloc = [&](size_t n) { float* p = w + o; o += n; return p; };
    float* xn    = alloc((size_t)Bsz * Dsz);
    float* Sb    = alloc((size_t)Bsz * Bsz);   // S -> A -> A_norm in place
    float* Kb0   = alloc((size_t)Bsz * Bsz);
    float* Kb1   = alloc((size_t)Bsz * Bsz);
    float* H1    = alloc((size_t)Bsz * Dsz);
    float* H2    = alloc((size_t)Bsz * Dsz);
    float* Zb    = alloc((size_t)Bsz * Dsz);
    float* Zt    = alloc((size_t)Bsz * Dsz);
    float* P0    = alloc((size_t)Bsz * Psz);
    float* P1    = alloc((size_t)Bsz * Psz);
    float* Pt    = alloc((size_t)Bsz * Psz);
    unsigned* mIntra = (unsigned*)alloc((size_t)Bsz * 64);
    unsigned* mInter = (unsigned*)alloc((size_t)Bsz * 64);
    float* deg   = alloc(Bsz);
    float* dinv  = alloc(Bsz);
    float* dinvI = alloc(Bsz);
    float* partials = alloc(1024);
    float* scal  = alloc(8);

    init_scalars_kernel<<<1, 32, 0, stream>>>(scal);

    const float alphas[3] = {1.0f, 1.1f, 1.2f};
    const float betas [3] = {1.0f, 0.9f, 0.8f};
    float* Kprev = Kb0; float* Kcur = Kb1;
    float* Pprev = P0;  float* Pcur = P1;

    for (int i = 0; i < Lsz; ++i) {
        const float* X = feats + (size_t)i * Bsz * Dsz;

        // xn = l2norm(X); S = clip(xn xn^T)
        l2norm_rows_kernel<<<Bsz, 256, 0, stream>>>(X, xn, Dsz);
        gemm(xn, xn, Sb, Bsz, Bsz, Dsz, Dsz, Dsz, Bsz, 1, stream);
        clip_kernel<<<4096, 256, 0, stream>>>(Sb, (size_t)Bsz * Bsz);

        // knn masks, A construction, symmetric normalization
        topk_mask_kernel<<<Bsz, 256, 0, stream>>>(Sb, labels, mIntra, 1);
        topk_mask_kernel<<<Bsz, 256, 0, stream>>>(Sb, labels, mInter, 0);
        build_A_kernel<<<Bsz, 256, 0, stream>>>(Sb, mIntra, mInter, deg, alphas[i], betas[i]);
        dinv_kernel<<<8, 256, 0, stream>>>(deg, dinv);
        normsym_kernel<<<4096, 256, 0, stream>>>(Sb, dinv);

        // K = A_norm @ A_norm   (T_DIFF = 2)
        gemm(Sb, Sb, Kcur, Bsz, Bsz, Bsz, Bsz, Bsz, Bsz, 0, stream);

        // GAM: Z = A @ fc1(X) -> BN -> relu -> A @ fc2 -> + X
        gemm(X, fc1 + (size_t)i * Dsz * Dsz, H1, Bsz, Dsz, Dsz, Dsz, Dsz, Dsz, 1, stream);
        gemm(Sb, H1, Zt, Bsz, Dsz, Bsz, Bsz, Dsz, Dsz, 0, stream);
        bn_relu_kernel<<<Dsz, 256, 0, stream>>>(Zt, Zb, gamma + (size_t)i * Dsz, bbeta + (size_t)i * Dsz);
        gemm(Zb, fc2 + (size_t)i * Dsz * Dsz, H2, Bsz, Dsz, Dsz, Dsz, Dsz, Dsz, 1, stream);
        float* Zi = Zout + (size_t)i * Bsz * Dsz;
        gemm(Sb, H2, Zi, Bsz, Dsz, Bsz, Bsz, Dsz, Dsz, 0, stream);
        add_kernel<<<2048, 256, 0, stream>>>(Zi, X, (size_t)Bsz * Dsz);

        // projection for Z-alignment loss
        gemm(Zi, proj, Pt, Bsz, Psz, Dsz, Dsz, Dsz, Psz, 1, stream);
        l2norm_rows_kernel<<<Bsz, 256, 0, stream>>>(Pt, Pcur, Psz);

        if (i > 0) {
            sqdiff_partial_kernel<<<1024, 256, 0, stream>>>(Kprev, Kcur, partials, (size_t)Bsz * Bsz);
            finalize_add_kernel<<<1, 256, 0, stream>>>(partials, 1024, scal + 0,
                                                       1.0f / ((float)Bsz * (float)Bsz));
            sqdiff_partial_kernel<<<1024, 256, 0, stream>>>(Pprev, Pcur, partials, (size_t)Bsz * Psz);
            finalize_add_kernel<<<1, 256, 0, stream>>>(partials, 1024, scal + 1,
                                                       1.0f / ((float)Bsz * (float)Psz));
        }
        { float* tmp = Kprev; Kprev = Kcur; Kcur = tmp; }
        { float* tmp = Pprev; Pprev = Pcur; Pcur = tmp; }
    }

    // loss_idea vs analytically-normalized K_idea; Kprev holds K[L-1] after swap
    idea_dinv_kernel<<<8, 256, 0, stream>>>(labels, dinvI);
    idea_partial_kernel<<<1024, 256, 0, stream>>>(Kprev, labels, dinvI, partials);
    finalize_add_kernel<<<1, 256, 0, stream>>>(partials, 1024, scal + 2,
                                               1.0f / ((float)Bsz * (float)Bsz));

    write_losses_kernel<<<1, 32, 0, stream>>>(scal, Zout + (size_t)Lsz * Bsz * Dsz);
}